// MTGODE_50629074485267
// MI455X (gfx1250) — compile-verified
//
#include <hip/hip_runtime.h>
#include <hip/hip_bf16.h>
#include <math.h>

// ---------------------------------------------------------------------------
// MTGODE on gfx1250.  Key ideas:
//  * cgp's 4 propagation steps are linear: fold into M^4 (M = 0.75I + 0.25*Anorm),
//    precomputed once with WMMA GEMMs.
//  * 1x1 conv commutes with node matmul -> channel-mix f first (wave shuffles),
//    then ONE dual-source fp32 WMMA GEMM per outer iteration:
//        P = M1^4 * (g1_w . f)  +  M2^4 * (g2_w . f)
//  * fp32 kept end-to-end via V_WMMA_F32_16X16X4_F32 (exact, matrix-pipe).
//  * GEMM inner loop is software-pipelined (double-buffered fragments) so the
//    compiler can use a relaxed s_wait_loadcnt instead of a full drain.
// ---------------------------------------------------------------------------

typedef float v2f __attribute__((ext_vector_type(2)));
typedef float v8f __attribute__((ext_vector_type(8)));

#define NNODE 512
#define BATCH 16
#define CONVC 32
#define RFLEN 19

// ---------------- adjacency row sums (adj and adj^T) -----------------------
__global__ void rowsums_kernel(const float* __restrict__ adj,
                               float* __restrict__ r1, float* __restrict__ r2) {
  __shared__ float s1[256], s2[256];
  int v = blockIdx.x;
  float a1 = 0.f, a2 = 0.f;
  for (int w = threadIdx.x; w < NNODE; w += 256) {
    a1 += adj[v * NNODE + w];      // row sum of adj
    a2 += adj[w * NNODE + v];      // row sum of adj^T
  }
  s1[threadIdx.x] = a1; s2[threadIdx.x] = a2;
  __syncthreads();
  for (int s = 128; s > 0; s >>= 1) {
    if (threadIdx.x < (unsigned)s) {
      s1[threadIdx.x] += s1[threadIdx.x + s];
      s2[threadIdx.x] += s2[threadIdx.x + s];
    }
    __syncthreads();
  }
  if (threadIdx.x == 0) { r1[v] = s1[0] + 1.0f; r2[v] = s2[0] + 1.0f; }
}

// ---------------- build M = 0.75 I + 0.25 * normalized(adj + I) ------------
__global__ void build_m_kernel(const float* __restrict__ adj,
                               const float* __restrict__ r1,
                               const float* __restrict__ r2,
                               float* __restrict__ M1, float* __restrict__ M2) {
  int idx = blockIdx.x * blockDim.x + threadIdx.x;
  if (idx >= NNODE * NNODE) return;
  int v = idx >> 9, w = idx & (NNODE - 1);
  float d = (v == w) ? 1.0f : 0.0f;
  M1[idx] = 0.75f * d + 0.25f * (adj[v * NNODE + w] + d) / r1[v];
  M2[idx] = 0.75f * d + 0.25f * (adj[w * NNODE + v] + d) / r2[v];
}

// ---------------- start 1x1 conv: CIN=2 -> CONV=32 -------------------------
__global__ void start_conv_kernel(const float* __restrict__ x,
                                  const float* __restrict__ w,
                                  const float* __restrict__ bias,
                                  float* __restrict__ state) {
  const int TOT = BATCH * CONVC * NNODE * RFLEN;
  int idx = blockIdx.x * blockDim.x + threadIdx.x;
  if (idx >= TOT) return;
  int l = idx % RFLEN;
  int n = (idx / RFLEN) & (NNODE - 1);
  int c = (idx / (RFLEN * NNODE)) & (CONVC - 1);
  int b = idx / (RFLEN * NNODE * CONVC);
  float x0 = x[((b * 2 + 0) * NNODE + n) * RFLEN + l];
  float x1 = x[((b * 2 + 1) * NNODE + n) * RFLEN + l];
  state[idx] = w[c * 2 + 0] * x0 + w[c * 2 + 1] * x1 + bias[c];
}

// ---------------- fp32 WMMA GEMM fragments ---------------------------------
struct Frag {
  v2f a, b0, b1, b2, b3;
};

__device__ __forceinline__ Frag load_frag(const float* __restrict__ Abase,
                                          const float* __restrict__ Bbase,
                                          int k, int ncols) {
  Frag f;
  f.a = *(const v2f*)(Abase + k);
  const float* Br0 = Bbase + (size_t)k * ncols;
  const float* Br1 = Br0 + ncols;
  f.b0 = v2f{Br0[0],  Br1[0]};
  f.b1 = v2f{Br0[16], Br1[16]};
  f.b2 = v2f{Br0[32], Br1[32]};
  f.b3 = v2f{Br0[48], Br1[48]};
  return f;
}

__device__ __forceinline__ void mma_frag(const Frag& f, v8f& acc0, v8f& acc1,
                                         v8f& acc2, v8f& acc3) {
  acc0 = __builtin_amdgcn_wmma_f32_16x16x4_f32(false, f.a, false, f.b0, (short)0, acc0, false, false);
  acc1 = __builtin_amdgcn_wmma_f32_16x16x4_f32(false, f.a, false, f.b1, (short)0, acc1, false, false);
  acc2 = __builtin_amdgcn_wmma_f32_16x16x4_f32(false, f.a, false, f.b2, (short)0, acc2, false, false);
  acc3 = __builtin_amdgcn_wmma_f32_16x16x4_f32(false, f.a, false, f.b3, (short)0, acc3, false, false);
}

// acc += A(16x512) * B(512x64), software-pipelined 2 deep.
__device__ __forceinline__ void gemm_pair(const float* __restrict__ A,
                                          const float* __restrict__ B,
                                          int rA, int kHalf, int col0, int cB,
                                          int ncols,
                                          v8f& acc0, v8f& acc1, v8f& acc2, v8f& acc3) {
  const float* Abase = A + (size_t)rA * NNODE + kHalf;
  const float* Bbase = B + (size_t)kHalf * ncols + col0 + cB;
  Frag cur = load_frag(Abase, Bbase, 0, ncols);
#pragma unroll 4
  for (int k = 0; k < NNODE - 4; k += 4) {
    Frag nxt = load_frag(Abase, Bbase, k + 4, ncols);   // prefetch next gen
    mma_frag(cur, acc0, acc1, acc2, acc3);              // consume current gen
    cur = nxt;
  }
  mma_frag(cur, acc0, acc1, acc2, acc3);
}

// P = A1*B1 (+ A2*B2), K = 512 per pair.
// A: 512x512 row-major.  B: 512 x ncols row-major.  P: 512 x ncols.
// grid = (ncols/256, 512/32); block = 256 (8 waves: 2 row-tiles x 4 col-grps)
template <bool DUAL>
__global__ __launch_bounds__(256) void gemm512_wmma_kernel(
    const float* __restrict__ A1, const float* __restrict__ B1,
    const float* __restrict__ A2, const float* __restrict__ B2,
    float* __restrict__ P, int ncols) {
  int lane = threadIdx.x & 31;
  int wid  = threadIdx.x >> 5;
  int row0 = blockIdx.y * 32 + (wid & 1) * 16;
  int col0 = blockIdx.x * 256 + (wid >> 1) * 64;
  int rA    = row0 + (lane & 15);          // A fragment row
  int kHalf = (lane >> 4) << 1;            // lanes 0-15: K0/K1, 16-31: K2/K3
  int cB    = lane & 15;                   // B fragment col

  v8f acc0 = {}, acc1 = {}, acc2 = {}, acc3 = {};

  gemm_pair(A1, B1, rA, kHalf, col0, cB, ncols, acc0, acc1, acc2, acc3);
  if (DUAL)
    gemm_pair(A2, B2, rA, kHalf, col0, cB, ncols, acc0, acc1, acc2, acc3);

  int rowBase = row0 + (lane >> 4) * 8;
#pragma unroll
  for (int r = 0; r < 8; ++r) {
    float* Pp = P + (size_t)(rowBase + r) * ncols + col0 + cB;
    Pp[0]  = acc0[r];
    Pp[16] = acc1[r];
    Pp[32] = acc2[r];
    Pp[48] = acc3[r];
  }
}

// ---------------- gated inception + channel pre-mix ------------------------
struct IncW {
  const float* w1[4]; const float* b1[4];
  const float* w2[4]; const float* b2[4];
  const float* g1w;   const float* g2w;
};

// one wave per (b,n,l); lane = channel c.  Writes F1/F2 in node-major GEMM
// layout: F[n][ (b*32+c)*Lf + l ],   ncols = 512*Lf.
__global__ __launch_bounds__(256) void inception_gate_kernel(
    const float* __restrict__ state, IncW IW,
    float* __restrict__ F1, float* __restrict__ F2, int seq) {
  const int Lf = seq - 6;
  const int off = RFLEN - seq;
  const int ncols = NNODE * Lf;  // == BATCH*CONVC*Lf
  int lane = threadIdx.x & 31;
  int wid  = threadIdx.x >> 5;
  int tuple = blockIdx.x * 8 + wid;
  int l = tuple % Lf;
  int n = (tuple / Lf) & (NNODE - 1);
  int b = tuple / (Lf * NNODE);

  const int c = lane;
  const int branch = c >> 3, co = c & 7;
  const int ksetA[4] = {2, 3, 6, 7};
  const int k = ksetA[branch];
  const float* w1 = IW.w1[branch] + co * CONVC * k;
  const float* w2 = IW.w2[branch] + co * CONVC * k;
  float y1 = IW.b1[branch][co];
  float y2 = IW.b2[branch][co];

  for (int ci = 0; ci < CONVC; ++ci) {
    const float* sp = state + (((b * CONVC + ci) * NNODE + n) * RFLEN + off + l);
    const float* ww1 = w1 + ci * k;
    const float* ww2 = w2 + ci * k;
#pragma unroll
    for (int p = 0; p < 7; ++p) {           // shared taps: xs[l+p], p=0..6
      float xs = sp[p];
      int t = p - (7 - k);                  // tap index into width-k filter
      if (t >= 0) { y1 += ww1[t] * xs; y2 += ww2[t] * xs; }
    }
  }
  float f = tanhf(y1) * (1.0f / (1.0f + __expf(-y2)));

  // channel pre-mix (commutes with node matmul): f1 = g1_w . f, f2 = g2_w . f
  float f1 = 0.f, f2 = 0.f;
#pragma unroll
  for (int cc = 0; cc < CONVC; ++cc) {
    float fc = __shfl(f, cc, 32);
    f1 += IW.g1w[c * CONVC + cc] * fc;
    f2 += IW.g2w[c * CONVC + cc] * fc;
  }
  int col = (b * CONVC + c) * Lf + l;
  F1[(size_t)n * ncols + col] = f1;
  F2[(size_t)n * ncols + col] = f2;
}

// ---------------- residual state update ------------------------------------
__global__ void apply_update_kernel(float* __restrict__ state,
                                    const float* __restrict__ P,
                                    const float* __restrict__ g1b,
                                    const float* __restrict__ g2b, int Lf) {
  const int tot = BATCH * CONVC * NNODE * Lf;
  int idx = blockIdx.x * blockDim.x + threadIdx.x;
  if (idx >= tot) return;
  int l = idx % Lf;
  int r = idx / Lf;
  int v = r & (NNODE - 1); r >>= 9;
  int co = r & (CONVC - 1);
  int b = r / CONVC;
  int ncols = NNODE * Lf;
  int col = (b * CONVC + co) * Lf + l;
  float g = P[(size_t)v * ncols + col] + g1b[co] + g2b[co];
  state[((b * CONVC + co) * NNODE + v) * RFLEN + (RFLEN - Lf + l)] += 0.4f * g;
}

// ---------------- per-batch layernorm stats over last timestep -------------
__global__ void batch_stats_kernel(const float* __restrict__ state,
                                   float* __restrict__ stats) {
  __shared__ float ss[256], sq[256];
  int b = blockIdx.x;
  float s = 0.f, q = 0.f;
  for (int i = threadIdx.x; i < CONVC * NNODE; i += 256) {
    int c = i >> 9, n = i & (NNODE - 1);
    float v = state[((b * CONVC + c) * NNODE + n) * RFLEN + (RFLEN - 1)];
    s += v; q += v * v;
  }
  ss[threadIdx.x] = s; sq[threadIdx.x] = q;
  __syncthreads();
  for (int st = 128; st > 0; st >>= 1) {
    if (threadIdx.x < (unsigned)st) {
      ss[threadIdx.x] += ss[threadIdx.x + st];
      sq[threadIdx.x] += sq[threadIdx.x + st];
    }
    __syncthreads();
  }
  if (threadIdx.x == 0) {
    const float inv = 1.0f / (CONVC * NNODE);
    float mu  = ss[0] * inv;
    float var = sq[0] * inv - mu * mu;
    stats[b * 2]     = mu;
    stats[b * 2 + 1] = rsqrtf(var + 1e-5f);
  }
}

// ---------------- MLP head: 32 -> 64 -> 128 -> 12 --------------------------
__global__ __launch_bounds__(128) void head_kernel(
    const float* __restrict__ state, const float* __restrict__ stats,
    const float* __restrict__ e0w, const float* __restrict__ e0b,
    const float* __restrict__ e1w, const float* __restrict__ e1b,
    const float* __restrict__ e2w, const float* __restrict__ e2b,
    float* __restrict__ out) {
  __shared__ float hn[32], a0[64], a1[128];
  int b = blockIdx.x >> 9;
  int n = blockIdx.x & (NNODE - 1);
  int t = threadIdx.x;
  float mu = stats[b * 2], istd = stats[b * 2 + 1];
  if (t < 32)
    hn[t] = (state[((b * CONVC + t) * NNODE + n) * RFLEN + (RFLEN - 1)] - mu) * istd;
  __syncthreads();
  if (t < 64) {
    float a = e0b[t];
#pragma unroll
    for (int c = 0; c < 32; ++c) a += e0w[t * 32 + c] * hn[c];
    a0[t] = fmaxf(a, 0.f);
  }
  __syncthreads();
  {
    float a = e1b[t];
#pragma unroll
    for (int j = 0; j < 64; ++j) a += e1w[t * 64 + j] * a0[j];
    a1[t] = fmaxf(a, 0.f);
  }
  __syncthreads();
  if (t < 12) {
    float a = e2b[t];
#pragma unroll
    for (int j = 0; j < 128; ++j) a += e2w[t * 128 + j] * a1[j];
    out[(b * 12 + t) * NNODE + n] = a;
  }
}

// ---------------------------------------------------------------------------
extern "C" void kernel_launch(void* const* d_in, const int* in_sizes, int n_in,
                              void* d_out, int out_size, void* d_ws, size_t ws_size,
                              hipStream_t stream) {
  // Input order: x, adj, start_w, start_b, inc1_ws[4], inc1_bs[4],
  //              inc2_ws[4], inc2_bs[4], g1_w, g1_b, g2_w, g2_b,
  //              e0_w, e0_b, e1_w, e1_b, e2_w, e2_b
  const float* x      = (const float*)d_in[0];
  const float* adj    = (const float*)d_in[1];
  const float* startw = (const float*)d_in[2];
  const float* startb = (const float*)d_in[3];
  IncW IW;
  for (int i = 0; i < 4; ++i) {
    IW.w1[i] = (const float*)d_in[4 + i];
    IW.b1[i] = (const float*)d_in[8 + i];
    IW.w2[i] = (const float*)d_in[12 + i];
    IW.b2[i] = (const float*)d_in[16 + i];
  }
  const float* g1w = (const float*)d_in[20];
  const float* g1b = (const float*)d_in[21];
  const float* g2w = (const float*)d_in[22];
  const float* g2b = (const float*)d_in[23];
  const float* e0w = (const float*)d_in[24];
  const float* e0b = (const float*)d_in[25];
  const float* e1w = (const float*)d_in[26];
  const float* e1b = (const float*)d_in[27];
  const float* e2w = (const float*)d_in[28];
  const float* e2b = (const float*)d_in[29];
  IW.g1w = g1w; IW.g2w = g2w;

  // Workspace layout (floats)
  float* ws   = (float*)d_ws;
  float* M1   = ws;                    // 512*512
  float* M2   = M1 + NNODE * NNODE;
  float* T1   = M2 + NNODE * NNODE;
  float* T2   = T1 + NNODE * NNODE;
  float* r1   = T2 + NNODE * NNODE;    // 512
  float* r2   = r1 + NNODE;
  float* stats = r2 + NNODE;           // 32
  float* state = stats + 64;           // 16*32*512*19
  float* F1   = state + BATCH * CONVC * NNODE * RFLEN;
  float* F2   = F1 + (size_t)NNODE * NNODE * 13;
  float* P    = F2 + (size_t)NNODE * NNODE * 13;

  // 1) normalized propagation matrices M = 0.75I + 0.25*Anorm
  rowsums_kernel<<<NNODE, 256, 0, stream>>>(adj, r1, r2);
  build_m_kernel<<<(NNODE * NNODE + 255) / 256, 256, 0, stream>>>(adj, r1, r2, M1, M2);

  // 2) start conv
  const int stateN = BATCH * CONVC * NNODE * RFLEN;
  start_conv_kernel<<<(stateN + 255) / 256, 256, 0, stream>>>(x, startw, startb, state);

  // 3) M^4 via two squarings (WMMA GEMMs, ncols = 512)
  dim3 gPow(NNODE / 256, NNODE / 32);
  gemm512_wmma_kernel<false><<<gPow, 256, 0, stream>>>(M1, M1, nullptr, nullptr, T1, NNODE);
  gemm512_wmma_kernel<false><<<gPow, 256, 0, stream>>>(M2, M2, nullptr, nullptr, T2, NNODE);
  gemm512_wmma_kernel<false><<<gPow, 256, 0, stream>>>(T1, T1, nullptr, nullptr, M1, NNODE);
  gemm512_wmma_kernel<false><<<gPow, 256, 0, stream>>>(T2, T2, nullptr, nullptr, M2, NNODE);

  // 4) NFE1 outer iterations
  const int seqs[3] = {19, 13, 7};
  for (int it = 0; it < 3; ++it) {
    int seq = seqs[it];
    int Lf = seq - 6;
    int ncols = NNODE * Lf;
    int tuples = BATCH * NNODE * Lf;
    inception_gate_kernel<<<tuples / 8, 256, 0, stream>>>(state, IW, F1, F2, seq);
    dim3 gG(ncols / 256, NNODE / 32);
    gemm512_wmma_kernel<true><<<gG, 256, 0, stream>>>(M1, F1, M2, F2, P, ncols);
    int upd = BATCH * CONVC * NNODE * Lf;
    apply_update_kernel<<<(upd + 255) / 256, 256, 0, stream>>>(state, P, g1b, g2b, Lf);
  }

  // 5) layernorm + MLP head
  batch_stats_kernel<<<BATCH, 256, 0, stream>>>(state, stats);
  head_kernel<<<BATCH * NNODE, 128, 0, stream>>>(state, stats, e0w, e0b, e1w, e1b,
                                                 e2w, e2b, (float*)d_out);
}